// SparseMultiHeadAttention_75333726372244
// MI455X (gfx1250) — compile-verified
//
#include <hip/hip_runtime.h>

typedef float v2f __attribute__((ext_vector_type(2)));
typedef float v8f __attribute__((ext_vector_type(8)));

#define HH   8
#define DMD  512
#define DKV  64
#define BB   2
#define LL   2048
#define NN   16      /* H*B */
#define TT   50
#define PPHI 20

// ---------------------------------------------------------------- WMMA helper
__device__ __forceinline__ v8f wmma4(v2f a, v2f b, v8f c) {
  // V_WMMA_F32_16X16X4_F32 : D(16x16,f32) = A(16x4,f32) x B(4x16,f32) + C
  return __builtin_amdgcn_wmma_f32_16x16x4_f32(false, a, false, b, (short)0, c,
                                               false, false);
}

// ------------------------------------------------- threefry2x32 (JAX-compatible)
__device__ __forceinline__ void threefry2x32(unsigned k0, unsigned k1,
                                             unsigned x0, unsigned x1,
                                             unsigned& o0, unsigned& o1) {
  const unsigned k2 = k0 ^ k1 ^ 0x1BD11BDAu;
  x0 += k0; x1 += k1;
#define TFR(r) { x0 += x1; x1 = (x1 << (r)) | (x1 >> (32 - (r))); x1 ^= x0; }
  TFR(13) TFR(15) TFR(26) TFR(6)   x0 += k1; x1 += k2 + 1u;
  TFR(17) TFR(29) TFR(16) TFR(24)  x0 += k2; x1 += k0 + 2u;
  TFR(13) TFR(15) TFR(26) TFR(6)   x0 += k0; x1 += k1 + 3u;
  TFR(17) TFR(29) TFR(16) TFR(24)  x0 += k1; x1 += k2 + 4u;
  TFR(13) TFR(15) TFR(26) TFR(6)   x0 += k2; x1 += k0 + 5u;
#undef TFR
  o0 = x0; o1 = x1;
}

// random_bits(key, 32, (dv,n,T,PHI)): iota split into halves (N = 1,024,000)
__device__ __forceinline__ unsigned rbits(unsigned k0, unsigned k1, unsigned f) {
  const unsigned HALF = 512000u;
  unsigned o0, o1;
  if (f < HALF) { threefry2x32(k0, k1, f, f + HALF, o0, o1); return o0; }
  threefry2x32(k0, k1, f - HALF, f, o0, o1); return o1;
}

// ---------------------------------------------------------------- projections
// Y = (X @ W^T + bias) * scale, written in (h*B+b, l, d) head-major layout.
__global__ __launch_bounds__(256) void proj_kernel(
    const float* __restrict__ Xq, const float* __restrict__ Xk,
    const float* __restrict__ Xv, const float* __restrict__ Wq,
    const float* __restrict__ Wk, const float* __restrict__ Wv,
    const float* __restrict__ bq, const float* __restrict__ bk,
    const float* __restrict__ bv, float* __restrict__ Yq,
    float* __restrict__ Yk, float* __restrict__ Yv, float qscale) {
  const float* X; const float* W; const float* bias; float* Y; float scale;
  if (blockIdx.z == 0)      { X = Xq; W = Wq; bias = bq; Y = Yq; scale = qscale; }
  else if (blockIdx.z == 1) { X = Xk; W = Wk; bias = bk; Y = Yk; scale = 1.f; }
  else                      { X = Xv; W = Wv; bias = bv; Y = Yv; scale = 1.f; }

  const int wave = threadIdx.x >> 5;
  const int lane = threadIdx.x & 31;
  const int lr = lane & 15, hi = lane >> 4;
  const int r0 = blockIdx.x << 4;                       // row tile (b*L + l)
  const int c0 = (((int)blockIdx.y << 3) + wave) << 4;  // col tile (h*64 + d)

  const float* xrow = X + (size_t)(r0 + lr) * DMD + 2 * hi;
  const float* wrow = W + (size_t)(c0 + lr) * DMD + 2 * hi;

  v8f acc = {};
#pragma unroll 8
  for (int c = 0; c < DMD / 4; ++c) {
    v2f a = *(const v2f*)(xrow + 4 * c);
    v2f b = *(const v2f*)(wrow + 4 * c);
    acc = wmma4(a, b, acc);
  }
  const int col = c0 + lr;
  const int h = col >> 6, d = col & 63;
  const float bb = bias[col];
#pragma unroll
  for (int j = 0; j < 8; ++j) {
    const int row = r0 + j + 8 * hi;
    const int b_ = row >> 11, l_ = row & (LL - 1);
    Y[((size_t)(h * BB + b_) * LL + l_) * DKV + d] = (acc[j] + bb) * scale;
  }
}

// -------------------------------------------- per-channel 16x16 Gram of vh rows
__global__ __launch_bounds__(256) void gram_kernel(const float* __restrict__ vh,
                                                   float* __restrict__ G) {
  const int i = blockIdx.x;           // 0..15
  const int c = threadIdx.x & 63;     // channel
  const int ms = threadIdx.x >> 6;    // m-slice 0..3
  float acc[16] = {};
  const int m0 = ms * (LL / 4);
  for (int m = m0; m < m0 + LL / 4; ++m) {
    const float vi = vh[((size_t)i * LL + m) * DKV + c];
#pragma unroll
    for (int j = 0; j < 16; ++j)
      acc[j] = fmaf(vi, vh[((size_t)j * LL + m) * DKV + c], acc[j]);
  }
  __shared__ float red[4][16][64];
#pragma unroll
  for (int j = 0; j < 16; ++j) red[ms][j][c] = acc[j];
  __syncthreads();
  if (ms == 0) {
#pragma unroll
    for (int j = 0; j < 16; ++j)
      G[(size_t)c * 256 + i * 16 + j] =
          red[0][j][c] + red[1][j][c] + red[2][j][c] + red[3][j][c];
  }
}

// ------------------------------------------ sim + threefry sampling -> data_score
__global__ __launch_bounds__(64) void dscore_kernel(const float* __restrict__ G,
                                                    float* __restrict__ ds) {
  const int i = blockIdx.x;    // n index
  const int c = threadIdx.x;   // channel
  __shared__ float simS[64][16];
  const float* Gc = G + (size_t)c * 256;
  const float sqi = Gc[i * 16 + i];
#pragma unroll
  for (int j = 0; j < 16; ++j) {
    const float d2 = sqi + Gc[j * 16 + j] - 2.f * Gc[i * 16 + j];
    simS[c][j] = 1.f / (1.f + sqrtf(fmaxf(d2, 0.f)));
  }
  // split(key(42)) -> k1=(k1a,k1b), k2=(k2a,k2b)
  unsigned k1a, k1b, k2a, k2b;
  threefry2x32(0u, 42u, 0u, 2u, k1a, k2a);
  threefry2x32(0u, 42u, 1u, 3u, k1b, k2b);
  float score = 0.f;
  const unsigned base = (unsigned)(c * 16 + i) * 1000u;
  for (int t = 0; t < TT; ++t) {
    float best = 0.f;
    for (int p = 0; p < PPHI; ++p) {
      const unsigned f = base + (unsigned)(t * PPHI + p);
      const unsigned hb = rbits(k1a, k1b, f);
      const unsigned lb = rbits(k2a, k2b, f);
      const unsigned idx = (hb % 15u + lb % 15u) % 15u;  // multiplier==1 for span 15
      best = fmaxf(best, simS[c][idx]);
    }
    score += best;
  }
  ds[i * 64 + c] = (float)TT / score;
}

// ------------------------------- scores + softmax + attn store (the big kernel)
__global__ __launch_bounds__(512) void attn_kernel(const float* __restrict__ qh,
                                                   const float* __restrict__ kh,
                                                   float* __restrict__ attn) {
  const int n = blockIdx.y;
  const int q0 = blockIdx.x << 5;          // 32 q-rows per block
  const int wave = threadIdx.x >> 5;       // 16 waves
  const int lane = threadIdx.x & 31;
  const int lr = lane & 15, hi = lane >> 4;

  v2f afr[2][16];
#pragma unroll
  for (int s = 0; s < 2; ++s) {
    const float* qrow = qh + ((size_t)n * LL + q0 + 16 * s + lr) * DKV + 2 * hi;
#pragma unroll
    for (int c = 0; c < 16; ++c) afr[s][c] = *(const v2f*)(qrow + 4 * c);
  }

  const int kstart = wave << 7;            // 128 k-columns per wave
  const float* kbase = kh + (size_t)n * LL * DKV;

  v8f acc[2][8];
#pragma unroll
  for (int t = 0; t < 8; ++t) {
    const float* krow = kbase + (size_t)(kstart + 16 * t + lr) * DKV + 2 * hi;
    if (t < 7) __builtin_prefetch(krow + 16 * DKV, 0, 0);
    v8f a0 = {}, a1 = {};
#pragma unroll
    for (int c = 0; c < 16; ++c) {
      const v2f b = *(const v2f*)(krow + 4 * c);
      a0 = wmma4(afr[0][c], b, a0);
      a1 = wmma4(afr[1][c], b, a1);
    }
    acc[0][t] = a0; acc[1][t] = a1;
  }

  __shared__ float wred[16][32];
  float m8[2][8], inv8[2][8];

  // row max: per-lane over tiles, shfl over 16-lane half, LDS over waves
#pragma unroll
  for (int s = 0; s < 2; ++s)
#pragma unroll
    for (int j = 0; j < 8; ++j) {
      float m = acc[s][0][j];
#pragma unroll
      for (int t = 1; t < 8; ++t) m = fmaxf(m, acc[s][t][j]);
#pragma unroll
      for (int off = 1; off < 16; off <<= 1) m = fmaxf(m, __shfl_xor(m, off, 32));
      if (lr == 0) wred[wave][16 * s + j + 8 * hi] = m;
    }
  __syncthreads();
#pragma unroll
  for (int s = 0; s < 2; ++s)
#pragma unroll
    for (int j = 0; j < 8; ++j) {
      float m = wred[0][16 * s + j + 8 * hi];
#pragma unroll
      for (int w = 1; w < 16; ++w) m = fmaxf(m, wred[w][16 * s + j + 8 * hi]);
      m8[s][j] = m;
    }
  __syncthreads();

  // exp + row sums
#pragma unroll
  for (int s = 0; s < 2; ++s)
#pragma unroll
    for (int j = 0; j < 8; ++j) {
      float sum = 0.f;
#pragma unroll
      for (int t = 0; t < 8; ++t) {
        const float e = __expf(acc[s][t][j] - m8[s][j]);
        acc[s][t][j] = e;
        sum += e;
      }
#pragma unroll
      for (int off = 1; off < 16; off <<= 1) sum += __shfl_xor(sum, off, 32);
      if (lr == 0) wred[wave][16 * s + j + 8 * hi] = sum;
    }
  __syncthreads();
#pragma unroll
  for (int s = 0; s < 2; ++s)
#pragma unroll
    for (int j = 0; j < 8; ++j) {
      float sum = 0.f;
#pragma unroll
      for (int w = 0; w < 16; ++w) sum += wred[w][16 * s + j + 8 * hi];
      inv8[s][j] = 1.f / sum;
    }

  // normalized attention -> HBM (roofline term: 268 MB)
#pragma unroll
  for (int s = 0; s < 2; ++s)
#pragma unroll
    for (int j = 0; j < 8; ++j) {
      const int row = q0 + 16 * s + j + 8 * hi;
      float* arow = attn + ((size_t)n * LL + row) * LL + kstart + lr;
      const float sc = inv8[s][j];
#pragma unroll
      for (int t = 0; t < 8; ++t) arow[16 * t] = acc[s][t][j] * sc;
    }
}

// ---------------------- z[n,l] = sum of masked attn columns (k<=n, parity even)
__global__ __launch_bounds__(256) void z_kernel(const float* __restrict__ attn,
                                                float* __restrict__ z) {
  const int tid = blockIdx.x * 256 + threadIdx.x;  // 16*2048
  const int n = tid >> 11, l = tid & (LL - 1);
  const float* row = attn + ((size_t)n * LL + l) * LL;
  float s = 0.f;
  for (int kk = (n & 1); kk <= n; kk += 2) s += row[kk];
  z[tid] = s;
}

// ---------------------------- M[n,c] = sum_d data_score[d,n] * wf[c, h*64 + d]
__global__ __launch_bounds__(256) void mmat_kernel(const float* __restrict__ ds,
                                                   const float* __restrict__ wf,
                                                   float* __restrict__ M) {
  const int tid = blockIdx.x * 256 + threadIdx.x;  // 16*512
  const int n = tid >> 9, c = tid & 511;
  const int h = n >> 1;
  const float* wrow = wf + (size_t)c * (HH * DKV) + h * DKV;
  const float* dsr = ds + n * 64;
  float s = 0.f;
#pragma unroll 16
  for (int d = 0; d < 64; ++d) s = fmaf(dsr[d], wrow[d], s);
  M[tid] = s;
}

// ---------------- out = LayerNorm( sum_h M[n,c]*z[n,l] + bf + residual(q) )
__global__ __launch_bounds__(256) void final_kernel(
    const float* __restrict__ q, const float* __restrict__ z,
    const float* __restrict__ M, const float* __restrict__ bf,
    const float* __restrict__ gamma, const float* __restrict__ beta,
    float* __restrict__ out) {
  const int r = blockIdx.x;                 // b*L + l
  const int b = r >> 11, l = r & (LL - 1);
  const int t = threadIdx.x;
  float zv[8];
#pragma unroll
  for (int h = 0; h < 8; ++h) zv[h] = z[(size_t)(h * BB + b) * LL + l];
  float x[2], lsum = 0.f, lsq = 0.f;
#pragma unroll
  for (int u = 0; u < 2; ++u) {
    const int c = t + 256 * u;
    float a = bf[c] + q[(size_t)r * DMD + c];
#pragma unroll
    for (int h = 0; h < 8; ++h)
      a = fmaf(M[(size_t)(h * BB + b) * DMD + c], zv[h], a);
    x[u] = a; lsum += a; lsq = fmaf(a, a, lsq);
  }
  __shared__ float s1[256], s2[256];
  s1[t] = lsum; s2[t] = lsq;
  __syncthreads();
  for (int off = 128; off > 0; off >>= 1) {
    if (t < off) { s1[t] += s1[t + off]; s2[t] += s2[t + off]; }
    __syncthreads();
  }
  const float mu = s1[0] * (1.f / DMD);
  const float var = s2[0] * (1.f / DMD) - mu * mu;
  const float inv = rsqrtf(var + 1e-5f);
#pragma unroll
  for (int u = 0; u < 2; ++u) {
    const int c = t + 256 * u;
    out[(size_t)r * DMD + c] = (x[u] - mu) * inv * gamma[c] + beta[c];
  }
}

// ------------------------------------------------------------------- launcher
extern "C" void kernel_launch(void* const* d_in, const int* in_sizes, int n_in,
                              void* d_out, int out_size, void* d_ws,
                              size_t ws_size, hipStream_t stream) {
  (void)in_sizes; (void)n_in; (void)out_size; (void)ws_size;
  const float* q  = (const float*)d_in[0];
  const float* k  = (const float*)d_in[1];
  const float* v  = (const float*)d_in[2];
  const float* wq = (const float*)d_in[3];
  const float* bq = (const float*)d_in[4];
  const float* wk = (const float*)d_in[5];
  const float* bk = (const float*)d_in[6];
  const float* wv = (const float*)d_in[7];
  const float* bv = (const float*)d_in[8];
  const float* wf = (const float*)d_in[9];
  const float* bf = (const float*)d_in[10];
  const float* gamma = (const float*)d_in[11];
  const float* beta  = (const float*)d_in[12];

  float* out  = (float*)d_out;
  float* attn = out + (size_t)BB * LL * DMD;   // outputs: out then attn

  float* ws = (float*)d_ws;                    // ~25.4 MB total
  float* qh = ws;
  float* kh = qh + (size_t)NN * LL * DKV;
  float* vh = kh + (size_t)NN * LL * DKV;
  float* G  = vh + (size_t)NN * LL * DKV;
  float* ds = G + 64 * 256;
  float* zb = ds + NN * 64;
  float* Mm = zb + NN * LL;

  const float qscale = 0.43527528164806206f;   // 1 / 64**0.2 (fold 1/TEMP into qh)

  proj_kernel<<<dim3(256, 4, 3), 256, 0, stream>>>(q, k, v, wq, wk, wv, bq, bk,
                                                   bv, qh, kh, vh, qscale);
  gram_kernel<<<16, 256, 0, stream>>>(vh, G);
  dscore_kernel<<<16, 64, 0, stream>>>(G, ds);
  attn_kernel<<<dim3(64, 16), 512, 0, stream>>>(qh, kh, attn);
  z_kernel<<<128, 256, 0, stream>>>(attn, zb);
  mmat_kernel<<<32, 256, 0, stream>>>(ds, wf, Mm);
  final_kernel<<<4096, 256, 0, stream>>>(q, zb, Mm, bf, gamma, beta, out);
}